// DWT3D_13237089206463
// MI455X (gfx1250) — compile-verified
//
#include <hip/hip_runtime.h>
#include <cstdint>

// 3-D Haar DWT, (2,32,64,128,128) fp32 -> low (2,32,32,64,64) + highs (2,32,7,32,64,64).
// Memory-bound: 512 MB traffic @ 23.3 TB/s ~= 22 us. Strategy: stream via the CDNA5
// async global->LDS engine (ASYNCcnt) with double buffering; math is a scaled 8-point
// Walsh-Hadamard butterfly; outputs leave through non-temporal stores.

#define TPB   256
#define ITERS 4

typedef float f2 __attribute__((ext_vector_type(2)));

// Geometry (all powers of two):
//  N = B*C = 64 volumes of (D=64,H=128,W=128); D2=32,H2=64,W2=64,W-pairs=32
//  plane vol = 1<<17 floats; low total = 1<<23 floats
//  work items (block-pairs) = 64*32*64*32 = 4,194,304
//  per-thread iteration stride = 1,048,576 work items = +16 volumes exactly
//    -> input byte delta  = 16<<22      = 0x4000000
//    -> low  float delta  = 16<<17
//    -> high float delta  = (16*7)<<17 = 112<<17

__device__ __forceinline__ void hadamard8(float v0, float v1, float v2, float v3,
                                          float v4, float v5, float v6, float v7,
                                          float* __restrict__ y) {
    float t0 = v0 + v1, t1 = v0 - v1;
    float t2 = v2 + v3, t3 = v2 - v3;
    float t4 = v4 + v5, t5 = v4 - v5;
    float t6 = v6 + v7, t7 = v6 - v7;
    float u0 = t0 + t2, u2 = t0 - t2;
    float u1 = t1 + t3, u3 = t1 - t3;
    float u4 = t4 + t6, u6 = t4 - t6;
    float u5 = t5 + t7, u7 = t5 - t7;
    const float k = 0.35355339059327373f;   // (1/sqrt(2))^3
    y[0] = (u0 + u4) * k;  y[4] = (u0 - u4) * k;
    y[1] = (u1 + u5) * k;  y[5] = (u1 - u5) * k;
    y[2] = (u2 + u6) * k;  y[6] = (u2 - u6) * k;
    y[3] = (u3 + u7) * k;  y[7] = (u3 - u7) * k;
}

__global__ __launch_bounds__(TPB) void haar3d_async_kernel(const float* __restrict__ x,
                                                           float* __restrict__ out)
{
    // Double-buffered per-thread staging: 2 * 256 threads * 64 B = 32 KB LDS.
    __shared__ __align__(16) float smem[2][TPB][16];

    const unsigned tid  = threadIdx.x;
    const unsigned gtid = blockIdx.x * TPB + tid;

    // ---- loop-invariant decompose (only n advances across iterations) ----
    const unsigned n   =  gtid >> 16;          // 0..15 at t=0
    const unsigned d0  = (gtid >> 11) & 31u;
    const unsigned h0  = (gtid >>  5) & 63u;
    const unsigned w0p =  gtid        & 31u;

    // input byte offset of row (2d0, 2h0, 4*w0p) inside volume stream
    const unsigned goff0 = ((n << 20) + (d0 << 15) + (h0 << 8) + (w0p << 2)) << 2;
    const uint64_t sx    = (uint64_t)x;        // SGPR base for saddr-form async loads

    // LDS byte addresses of the two staging buffers for this thread
    const uint32_t lbuf0 = (uint32_t)(uintptr_t)(&smem[0][tid][0]);
    const uint32_t lbuf1 = (uint32_t)(uintptr_t)(&smem[1][tid][0]);

    // output bases (advance by constants per iteration)
    const unsigned p = (d0 << 12) + (h0 << 6) + (w0p << 1);
    float* olow0 = out + (n << 17) + p;
    float* ohi0  = out + (1u << 23) + ((n * 7u) << 17) + p;

    // ---- async prefetch: 4 x b128 per thread, 512 B contiguous per wave-instruction ----
    auto prefetch = [&](int t, uint32_t lbuf) {
        const uint32_t off = goff0 + (uint32_t)t * 0x4000000u;
        asm volatile("global_load_async_to_lds_b128 %0, %1, %2"
                     :: "v"(lbuf       ), "v"(off         ), "s"(sx) : "memory");
        asm volatile("global_load_async_to_lds_b128 %0, %1, %2"
                     :: "v"(lbuf + 16u ), "v"(off +   512u), "s"(sx) : "memory");
        asm volatile("global_load_async_to_lds_b128 %0, %1, %2"
                     :: "v"(lbuf + 32u ), "v"(off + 65536u), "s"(sx) : "memory");
        asm volatile("global_load_async_to_lds_b128 %0, %1, %2"
                     :: "v"(lbuf + 48u ), "v"(off + 66048u), "s"(sx) : "memory");
    };

    // ---- consume one staged item: 2 butterflies + 8 NT float2 stores ----
    auto compute = [&](int t, int bf) {
        const float4* bp = reinterpret_cast<const float4*>(&smem[bf][tid][0]);
        const float4 r00 = bp[0];   // (2d0,   2h0  , 4w0p..+3)
        const float4 r01 = bp[1];   // (2d0,   2h0+1)
        const float4 r10 = bp[2];   // (2d0+1, 2h0  )
        const float4 r11 = bp[3];   // (2d0+1, 2h0+1)

        float yA[8], yB[8];
        hadamard8(r00.x, r00.y, r01.x, r01.y, r10.x, r10.y, r11.x, r11.y, yA);
        hadamard8(r00.z, r00.w, r01.z, r01.w, r10.z, r10.w, r11.z, r11.w, yB);

        float* ol = olow0 + (size_t)t * (16u << 17);
        float* oh = ohi0  + (size_t)t * (112u << 17);

        f2 v0 = { yA[0], yB[0] };
        __builtin_nontemporal_store(v0, reinterpret_cast<f2*>(ol));
#pragma unroll
        for (int s = 1; s < 8; ++s) {
            f2 v = { yA[s], yB[s] };
            // (s-1)<<17 floats = (s-1)*512KB: folds into the store's 24-bit IOFFSET
            __builtin_nontemporal_store(v, reinterpret_cast<f2*>(oh + ((unsigned)(s - 1) << 17)));
        }
    };

    // ---- software pipeline: prefetch t+1 while computing t ----
    prefetch(0, lbuf0);
#pragma unroll
    for (int t = 0; t < ITERS; ++t) {
        if (t + 1 < ITERS)
            prefetch(t + 1, ((t + 1) & 1) ? lbuf1 : lbuf0);
        if (t + 1 < ITERS) {
            // async loads retire in order: <=4 outstanding means the current
            // buffer's 4 copies have landed in LDS.
            asm volatile("s_wait_asynccnt 4" ::: "memory");
        } else {
            asm volatile("s_wait_asynccnt 0" ::: "memory");
        }
        compute(t, t & 1);
    }
}

extern "C" void kernel_launch(void* const* d_in, const int* in_sizes, int n_in,
                              void* d_out, int out_size, void* d_ws, size_t ws_size,
                              hipStream_t stream) {
    (void)in_sizes; (void)n_in; (void)d_ws; (void)ws_size; (void)out_size;
    const float* x = (const float*)d_in[0];   // (2,32,64,128,128) fp32
    // d_in[1] = Haar taps; they are the fixed scaled Hadamard, folded into the butterfly.
    float* out = (float*)d_out;               // low (1<<23 floats) ++ highs (7<<23 floats)

    const unsigned total_work = 64u * 32u * 64u * 32u;  // 4,194,304 block-pairs
    const unsigned threads    = total_work / ITERS;     // 1,048,576
    dim3 grid(threads / TPB);                           // 4096 blocks
    haar3d_async_kernel<<<grid, TPB, 0, stream>>>(x, out);
}